// CollisionLoss_25864293056654
// MI455X (gfx1250) — compile-verified
//
#include <hip/hip_runtime.h>

#define EGO_W   (1.85f + 0.5f)
#define EGO_L   (4.084f + 0.5f)
#define SOFT_M  1.5f
#define T_STEPS 12

typedef __attribute__((ext_vector_type(2))) float v2f;
typedef __attribute__((ext_vector_type(8))) float v8f;

__global__ void cl_init_kernel(float* acc, int* cnt) {
    if (threadIdx.x == 0) { acc[0] = 0.0f; cnt[0] = 0; }
}

__global__ __launch_bounds__(256) void cl_main_kernel(
    const float* __restrict__ pred_abs,    // (B, T, 2)
    const float* __restrict__ gt_bboxes,   // (B, N, 9)
    const int*   __restrict__ ego_masks,   // (B, T)
    const int*   __restrict__ agent_mask,  // (B, N)
    int N,
    float* __restrict__ loss_acc,
    int*   __restrict__ nvalid_acc)
{
    __shared__ float s_ex[T_STEPS];
    __shared__ float s_ey[T_STEPS];
    __shared__ float s_wavesum[8];

    const int b   = blockIdx.x;
    const int tid = threadIdx.x;

    // Broadcast ego trajectory; fold ego-valid mask by pushing invalid
    // timesteps far away so relu(thresh - |dist|) == 0 exactly.
    if (tid < T_STEPS) {
        const int  m  = ego_masks[b * T_STEPS + tid];
        const bool v  = (m >= 1);
        const float ex = pred_abs[(b * T_STEPS + tid) * 2 + 0];
        const float ey = pred_abs[(b * T_STEPS + tid) * 2 + 1];
        s_ex[tid] = v ? ex : 3.0e8f;
        s_ey[tid] = v ? ey : 3.0e8f;
    }
    if (tid == 0) {
        int c = 0;
        #pragma unroll
        for (int t = 0; t < T_STEPS; ++t)
            c += (ego_masks[b * T_STEPS + t] >= 1) ? 1 : 0;
        atomicAdd(nvalid_acc, c);
    }
    __syncthreads();

    float ex[T_STEPS], ey[T_STEPS];
    #pragma unroll
    for (int t = 0; t < T_STEPS; ++t) { ex[t] = s_ex[t]; ey[t] = s_ey[t]; }

    // Prefetch this thread's second agent record (next chunk) into caches.
    if (tid + 256 < N) {
        __builtin_prefetch(gt_bboxes + ((size_t)b * N + tid + 256) * 9, 0, 1);
    }

    const float cx_add = EGO_W * 0.5f + SOFT_M;
    const float cy_add = EGO_L * 0.5f + SOFT_M;

    float acc = 0.0f;
    for (int n = tid; n < N; n += 256) {
        const float* gb = gt_bboxes + ((size_t)b * N + n) * 9;
        const float cx = gb[0];
        const float cy = gb[1];
        const float ax = gb[3];
        const float ay = gb[4];
        const float am = (agent_mask[b * N + n] >= 1) ? 1.0f : 0.0f;

        const float thx = fmaf(ax, 0.5f, cx_add);
        const float thy = fmaf(ay, 0.5f, cy_add);

        float s = 0.0f;
        #pragma unroll
        for (int t = 0; t < T_STEPS; ++t) {
            const float px = fmaxf(thx - fabsf(ex[t] - cx), 0.0f);
            const float py = fmaxf(thy - fabsf(ey[t] - cy), 0.0f);
            s = fmaf(px, py, s);
        }
        acc = fmaf(s, am, acc);
    }

    // ---- Wave-level reduction via V_WMMA_F32_16X16X4_F32 (B = ones) ----
    // A layout (16x4 f32): lanes 0-15 -> row M=lane, VGPR0=K0, VGPR1=K1;
    //                      lanes 16-31 -> row M=lane-16, VGPR0=K2, VGPR1=K3.
    // With a.x = partial, a.y = 0 and B all-ones:
    //   D[m][n] = p_m + p_{m+16}  (independent of n).
    // C/D layout: lanes 0-15 hold M = vgpr (0..7); lanes 16-31 hold M = vgpr+8.
    // So sum of the 8 D VGPRs in lane 0  = sum(p_0..p_7)  + sum(p_16..p_23)
    //    sum of the 8 D VGPRs in lane 16 = sum(p_8..p_15) + sum(p_24..p_31).
    v2f a;  a.x = acc;  a.y = 0.0f;
    v2f bo; bo.x = 1.0f; bo.y = 1.0f;
    v8f c = {};
    c = __builtin_amdgcn_wmma_f32_16x16x4_f32(
            /*neg_a=*/false, a, /*neg_b=*/false, bo,
            /*c_mod=*/(short)0, c, /*reuse_a=*/false, /*reuse_b=*/false);

    float s8 = c[0] + c[1] + c[2] + c[3] + c[4] + c[5] + c[6] + c[7];
    const float wavesum = __shfl(s8, 0, 32) + __shfl(s8, 16, 32);

    const int wave = tid >> 5;
    const int lane = tid & 31;
    if (lane == 0) s_wavesum[wave] = wavesum;
    __syncthreads();

    if (tid == 0) {
        float tot = 0.0f;
        #pragma unroll
        for (int w = 0; w < 8; ++w) tot += s_wavesum[w];
        unsafeAtomicAdd(loss_acc, tot);   // global_atomic_add_f32
    }
}

__global__ void cl_finalize_kernel(const float* __restrict__ loss_acc,
                                   const int*   __restrict__ cnt,
                                   float* __restrict__ out)
{
    if (threadIdx.x == 0) {
        const float nv = fmaxf((float)cnt[0], 1.0f);
        out[0] = loss_acc[0] / nv;
    }
}

extern "C" void kernel_launch(void* const* d_in, const int* in_sizes, int n_in,
                              void* d_out, int out_size, void* d_ws, size_t ws_size,
                              hipStream_t stream) {
    const float* pred_abs   = (const float*)d_in[0];  // (B,T,2) f32
    const float* gt_bboxes  = (const float*)d_in[1];  // (B,N,9) f32
    const int*   ego_masks  = (const int*)  d_in[2];  // (B,T)   i32
    const int*   agent_mask = (const int*)  d_in[3];  // (B,N)   i32

    const int B = in_sizes[2] / T_STEPS;              // (B,T) ints
    const int N = in_sizes[3] / B;                    // (B,N) ints

    float* acc = (float*)d_ws;
    int*   cnt = (int*)((char*)d_ws + 8);

    cl_init_kernel<<<1, 1, 0, stream>>>(acc, cnt);
    cl_main_kernel<<<B, 256, 0, stream>>>(pred_abs, gt_bboxes, ego_masks,
                                          agent_mask, N, acc, cnt);
    cl_finalize_kernel<<<1, 1, 0, stream>>>(acc, cnt, (float*)d_out);
}